// DirectVoxGO_26817775796336
// MI455X (gfx1250) — compile-verified
//
#include <hip/hip_runtime.h>
#include <stdint.h>

// ---------------------------------------------------------------------------
// DirectVoxGO Raw2Alpha + Alphas2Weights (segmented exclusive transmittance)
//
// With INTERVAL = 0.5:  1-alpha = exp(-0.5*softplus(x)) = rsqrt(1 + exp(x))
// weights[i] = T[i]*alpha[i], T = per-ray exclusive cumprod of (1-alpha)
// ainv[r]    = per-ray total product (empty ray -> 1, matching exp(0))
//
// Memory-bound: ~200MB traffic -> ~8.6us floor @ 23.3 TB/s. One wave32 per
// ray (segment lengths ~Binomial, mean 128, std 11 -> near-perfect balance).
// CDNA5 paths: async global->LDS b128 double-buffered prefetch (ASYNCcnt),
// wave32 shuffle multiplicative scan, 4 elems/lane to amortize scan VALU.
// ---------------------------------------------------------------------------

// Kernel A: starts[r] = lower_bound(ray_id, r); starts[N] = M.
// Binary search touches ~O(N log M) mostly-L2-resident lines (<< 67MB stream).
__global__ void seg_starts_kernel(const int* __restrict__ ray_id, int M, int N,
                                  int* __restrict__ starts) {
    int r = blockIdx.x * blockDim.x + threadIdx.x;
    if (r > N) return;
    if (r == N) { starts[N] = M; return; }
    int lo = 0, hi = M;
    while (lo < hi) {
        int mid = (lo + hi) >> 1;
        if (ray_id[mid] < r) lo = mid + 1; else hi = mid;
    }
    starts[r] = lo;
}

// CDNA5 async copy: per-lane 16B gather from global into LDS (ASYNCcnt).
__device__ __forceinline__ void async_load_b128(unsigned lds_off, const float* gptr) {
    asm volatile("global_load_async_to_lds_b128 %0, %1, off"
                 :: "v"(lds_off), "v"((unsigned long long)(uintptr_t)gptr)
                 : "memory");
}

template <int WAVES>
__global__ __launch_bounds__(WAVES * 32)
void ray_scan_kernel(const float* __restrict__ density,
                     const float* __restrict__ shift,
                     const int* __restrict__ starts,
                     int N, int M,
                     float* __restrict__ weights,
                     float* __restrict__ ainv_last) {
    // 2 buffers x 128 f32 per wave, 16B-aligned for b128 LDS traffic
    __shared__ __align__(16) float buf[WAVES * 256];
    const int lane  = threadIdx.x & 31;
    const int wslot = threadIdx.x >> 5;
    const int ray   = blockIdx.x * WAVES + wslot;
    if (ray >= N) return;

    const int s  = starts[ray];
    const int e  = starts[ray + 1];
    const float sh = shift[0];
    const int nchunks = ((e - s) + 127) >> 7;          // 128 elems / chunk

    float* myb = &buf[wslot * 256];
    // flat address of __shared__: low 32 bits == wave-relative LDS byte offset
    const unsigned lds_base = (unsigned)(uintptr_t)myb;

    float carry = 1.0f;                                // running transmittance

    if (nchunks > 0) {
        {   // prefetch chunk 0: 512B per wave in one async b128 gather
            int cbase = s + lane * 4; if (cbase > M - 4) cbase = M - 4;
            async_load_b128(lds_base + (unsigned)lane * 16u, density + cbase);
        }
        for (int c = 0; c < nchunks; ++c) {
            const int nxt = c + 1;
            if (nxt < nchunks) {                       // prefetch chunk c+1
                int cbase = s + nxt * 128 + lane * 4; if (cbase > M - 4) cbase = M - 4;
                async_load_b128(lds_base + (unsigned)(((nxt & 1) << 9) + lane * 16u),
                                density + cbase);
                asm volatile("s_wait_asynccnt 0x1" ::: "memory");  // chunk c ready
            } else {
                asm volatile("s_wait_asynccnt 0x0" ::: "memory");
            }

            const int base = s + c * 128 + lane * 4;   // lane's first element
            const int rem  = e - base;                 // valid elems this lane
            const float* lb = &myb[((c & 1) << 7) + lane * 4];  // ds_load_b128

            float oma[4], alpha[4];
            #pragma unroll
            for (int k = 0; k < 4; ++k) {
                const float t = __builtin_amdgcn_exp2f((lb[k] + sh) * 1.4426950408889634f);
                float o = __builtin_amdgcn_rsqf(1.0f + t);   // 1 - alpha
                float a = 1.0f - o;
                if (k >= rem) { o = 1.0f; a = 0.0f; }        // identity for tail
                oma[k] = o; alpha[k] = a;
            }

            // lane-local product, then one wave32 multiplicative scan / 128 elems
            float p = (oma[0] * oma[1]) * (oma[2] * oma[3]);
            #pragma unroll
            for (int off = 1; off < 32; off <<= 1) {
                const float v = __shfl_up(p, off, 32);
                if (lane >= off) p *= v;
            }
            const float pex = __shfl_up(p, 1, 32);     // exclusive across lanes
            float T = (lane == 0) ? carry : carry * pex;

            #pragma unroll
            for (int k = 0; k < 4; ++k) {              // in-lane exclusive chain
                if (k < rem) weights[base + k] = T * alpha[k];
                T *= oma[k];
            }
            carry *= __shfl(p, 31, 32);                // fold chunk into carry
        }
    }
    if (lane == 0) ainv_last[ray] = carry;
}

extern "C" void kernel_launch(void* const* d_in, const int* in_sizes, int n_in,
                              void* d_out, int out_size, void* d_ws, size_t ws_size,
                              hipStream_t stream) {
    const float* density = (const float*)d_in[0];
    const int*   ray_id  = (const int*)d_in[1];
    const float* shift   = (const float*)d_in[2];
    // d_in[3] is N as a device scalar (unreadable during graph capture);
    // recover N from output layout: out = [weights(M) | alphainv_last(N)].
    const int M = in_sizes[0];
    const int N = out_size - M;

    int*   starts = (int*)d_ws;           // (N+1) ints of scratch
    float* out    = (float*)d_out;
    float* w      = out;                  // M weights
    float* ainv   = out + M;              // N final transmittances

    {
        const int threads = 256;
        const int blocks  = (N + 1 + threads - 1) / threads;
        seg_starts_kernel<<<blocks, threads, 0, stream>>>(ray_id, M, N, starts);
    }
    {
        constexpr int WAVES = 8;          // 256 threads = 8 wave32 rays / block
        const int blocks = (N + WAVES - 1) / WAVES;
        ray_scan_kernel<WAVES><<<blocks, WAVES * 32, 0, stream>>>(
            density, shift, starts, N, M, w, ainv);
    }
}